// GCNNet_56006373540375
// MI455X (gfx1250) — compile-verified
//
#include <hip/hip_runtime.h>

typedef float v2f __attribute__((ext_vector_type(2)));
typedef float v8f __attribute__((ext_vector_type(8)));

#define N_NODES_C 50000
#define N_EDGES_C 600000
#define IN_CH_C   64
#define HID_C     128
#define N_GRAPHS_C 64

#define PADW 136   // weight LDS row stride (136 % 64 == 8 -> K-half lane groups on disjoint banks)
#define LDA  132   // A-tile LDS row stride  (132 % 64 == 4 -> 16-row column read covers all banks)

// ---- CDNA5 async global->LDS copy (ASYNCcnt-tracked, bypasses VGPRs) -------
__device__ __forceinline__ void async_copy_b128(unsigned lds_off, const void* gptr) {
    unsigned long long ga = (unsigned long long)gptr;
    asm volatile("global_load_async_to_lds_b128 %0, %1, off"
                 :: "v"(lds_off), "v"(ga) : "memory");
}
__device__ __forceinline__ void wait_async_all() {
    asm volatile("s_wait_asynccnt 0x0" ::: "memory");
}

// ---------------------------------------------------------------------------
// Edge phase: msg = relu(x[src] + edge_attr @ el_w + el_b); agg[dst] += msg
// One thread per (edge, 4-channel group); float atomics resolve in L2
// (node arrays are 25.6MB << 192MB L2).
// ---------------------------------------------------------------------------
template<int C>
__global__ void gine_edge_kernel(const float* __restrict__ feat,
                                 const int*   __restrict__ srcI,
                                 const int*   __restrict__ dstI,
                                 const float* __restrict__ eattr,
                                 const float* __restrict__ elw,   // [3, C]
                                 const float* __restrict__ elb,   // [C]
                                 float*       __restrict__ agg,   // [N, C]
                                 int nedges) {
    const int groups = C / 4;
    int tid = blockIdx.x * blockDim.x + threadIdx.x;
    int e   = tid / groups;
    int c4  = (tid % groups) * 4;
    if (e >= nedges) return;

    float a0 = eattr[e * 3 + 0];
    float a1 = eattr[e * 3 + 1];
    float a2 = eattr[e * 3 + 2];
    int s = srcI[e];
    int d = dstI[e];

    float4 w0 = *(const float4*)(elw + 0 * C + c4);
    float4 w1 = *(const float4*)(elw + 1 * C + c4);
    float4 w2 = *(const float4*)(elw + 2 * C + c4);
    float4 bv = *(const float4*)(elb + c4);
    float4 xv = *(const float4*)(feat + (size_t)s * C + c4);

    float* ar = agg + (size_t)d * C + c4;

    float m0 = xv.x + a0 * w0.x + a1 * w1.x + a2 * w2.x + bv.x;
    float m1 = xv.y + a0 * w0.y + a1 * w1.y + a2 * w2.y + bv.y;
    float m2 = xv.z + a0 * w0.z + a1 * w1.z + a2 * w2.z + bv.z;
    float m3 = xv.w + a0 * w0.w + a1 * w1.w + a2 * w2.w + bv.w;
    m0 = m0 > 0.f ? m0 : 0.f;
    m1 = m1 > 0.f ? m1 : 0.f;
    m2 = m2 > 0.f ? m2 : 0.f;
    m3 = m3 > 0.f ? m3 : 0.f;
    atomicAdd(ar + 0, m0);
    atomicAdd(ar + 1, m1);
    atomicAdd(ar + 2, m2);
    atomicAdd(ar + 3, m3);
}

// ---------------------------------------------------------------------------
// Node phase (fused MLP, fp32 WMMA 16x16x4):
//   h   = x + agg
//   out = relu( relu(h @ wa + ba) @ wb + bb )
//
// 256 threads = 8 waves; each wave owns one 16-node M-tile (block = 128 rows).
// Weights are staged once per block into LDS via global_load_async_to_lds_b128
// and shared by all 8 waves; WMMA fragments then come entirely from LDS.
//
// ISA layouts (CDNA5 ISA):
//   A 16x4 f32 : lanes 0-15 = M, VGPR0/1 = K{0,1} (lanes 0-15) / K{2,3} (16-31)
//   B 4x16 f32 : lanes 0-15 = N, same K striping
//   C/D 16x16  : lane%16 = N, VGPR r = M=r (lanes 0-15) / M=r+8 (lanes 16-31)
// ---------------------------------------------------------------------------
template<int INCH>
__global__ __launch_bounds__(256)
void gine_node_mlp_kernel(const float* __restrict__ xin,
                          const float* __restrict__ agg,
                          const float* __restrict__ wa,  // [INCH,128]
                          const float* __restrict__ ba,  // [128]
                          const float* __restrict__ wb,  // [128,128]
                          const float* __restrict__ bb,  // [128]
                          float*       __restrict__ out, // [N,128]
                          int nnodes) {
    __shared__ __align__(16) float sWa[INCH  * PADW];
    __shared__ __align__(16) float sWb[HID_C * PADW];
    __shared__ __align__(16) float sT[8 * 16 * LDA];    // per-wave A / t tiles

    const int lane  = threadIdx.x & 31;
    const int wave  = threadIdx.x >> 5;
    const int m0    = blockIdx.x * 128 + wave * 16;
    const int mrow  = lane & 15;
    const int khalf = (lane >> 4) * 2;      // 0 or 2
    float* sA = sT + wave * 16 * LDA;

    // ---- async-stage weights into padded LDS (block-wide, VGPR-bypassing) ----
    {
        const unsigned baseWa = (unsigned)(size_t)sWa;
        for (int ch = threadIdx.x; ch < INCH * 32; ch += 256) {
            int row = ch >> 5, c4 = (ch & 31) * 4;
            async_copy_b128(baseWa + (unsigned)(row * PADW + c4) * 4u,
                            wa + row * HID_C + c4);
        }
        const unsigned baseWb = (unsigned)(size_t)sWb;
        for (int ch = threadIdx.x; ch < HID_C * 32; ch += 256) {
            int row = ch >> 5, c4 = (ch & 31) * 4;
            async_copy_b128(baseWb + (unsigned)(row * PADW + c4) * 4u,
                            wb + row * HID_C + c4);
        }
    }

    // ---- stage A = x + agg (16 x INCH) into this wave's LDS tile ----
    for (int i = lane; i < 16 * INCH; i += 32) {
        int r = i / INCH, c = i % INCH;
        int node = m0 + r;
        float v = 0.f;
        if (node < nnodes)
            v = xin[(size_t)node * INCH + c] + agg[(size_t)node * INCH + c];
        sA[r * LDA + c] = v;
    }
    wait_async_all();
    __syncthreads();

    v8f vzero = {};
    v8f acc[8];
#pragma unroll
    for (int nt = 0; nt < 8; ++nt) acc[nt] = vzero;

    // ---- GEMM1: (16 x INCH) @ (INCH x 128), all fragments from LDS ----
#pragma unroll 1
    for (int k0 = 0; k0 < INCH; k0 += 4) {
        v2f afrag;
        afrag.x = sA[mrow * LDA + k0 + khalf + 0];
        afrag.y = sA[mrow * LDA + k0 + khalf + 1];
#pragma unroll
        for (int nt = 0; nt < 8; ++nt) {
            int n = nt * 16 + mrow;
            v2f bfrag;
            bfrag.x = sWa[(k0 + khalf + 0) * PADW + n];
            bfrag.y = sWa[(k0 + khalf + 1) * PADW + n];
            acc[nt] = __builtin_amdgcn_wmma_f32_16x16x4_f32(
                false, afrag, false, bfrag, (short)0, acc[nt], false, false);
        }
    }

    // ---- epilogue 1: bias + relu, re-layout through LDS ----
    __syncthreads();
#pragma unroll
    for (int nt = 0; nt < 8; ++nt) {
#pragma unroll
        for (int r = 0; r < 8; ++r) {
            int row = (lane < 16) ? r : (r + 8);
            int col = nt * 16 + mrow;
            float v = acc[nt][r] + ba[col];
            v = v > 0.f ? v : 0.f;
            sA[row * LDA + col] = v;
        }
    }
    __syncthreads();

    // ---- GEMM2: (16 x 128) @ (128 x 128) ----
#pragma unroll
    for (int nt = 0; nt < 8; ++nt) acc[nt] = vzero;
#pragma unroll 1
    for (int k0 = 0; k0 < HID_C; k0 += 4) {
        v2f afrag;
        afrag.x = sA[mrow * LDA + k0 + khalf + 0];
        afrag.y = sA[mrow * LDA + k0 + khalf + 1];
#pragma unroll
        for (int nt = 0; nt < 8; ++nt) {
            int n = nt * 16 + mrow;
            v2f bfrag;
            bfrag.x = sWb[(k0 + khalf + 0) * PADW + n];
            bfrag.y = sWb[(k0 + khalf + 1) * PADW + n];
            acc[nt] = __builtin_amdgcn_wmma_f32_16x16x4_f32(
                false, afrag, false, bfrag, (short)0, acc[nt], false, false);
        }
    }

    // ---- epilogue 2: bias + relu, store ----
#pragma unroll
    for (int nt = 0; nt < 8; ++nt) {
#pragma unroll
        for (int r = 0; r < 8; ++r) {
            int row = (lane < 16) ? r : (r + 8);
            int col = nt * 16 + mrow;
            int node = m0 + row;
            if (node < nnodes) {
                float v = acc[nt][r] + bb[col];
                v = v > 0.f ? v : 0.f;
                out[(size_t)node * HID_C + col] = v;
            }
        }
    }
}

// ---------------------------------------------------------------------------
// Global mean pool: per-graph sums + counts via atomics, then divide.
// ---------------------------------------------------------------------------
__global__ void pool_sum_kernel(const float* __restrict__ h,
                                const int*   __restrict__ batch,
                                float* __restrict__ sums,   // [G,128]
                                float* __restrict__ cnt,    // [G]
                                int nnodes) {
    const int groups = HID_C / 4;
    int tid  = blockIdx.x * blockDim.x + threadIdx.x;
    int node = tid / groups;
    int c4   = (tid % groups) * 4;
    if (node >= nnodes) return;
    int g = batch[node];
    float4 hv = *(const float4*)(h + (size_t)node * HID_C + c4);
    float* sr = sums + (size_t)g * HID_C + c4;
    atomicAdd(sr + 0, hv.x);
    atomicAdd(sr + 1, hv.y);
    atomicAdd(sr + 2, hv.z);
    atomicAdd(sr + 3, hv.w);
    if (c4 == 0) atomicAdd(&cnt[g], 1.0f);
}

__global__ void pool_div_kernel(const float* __restrict__ sums,
                                const float* __restrict__ cnt,
                                float* __restrict__ outp) {
    int i = blockIdx.x * blockDim.x + threadIdx.x;
    if (i < N_GRAPHS_C * HID_C) {
        float c = cnt[i / HID_C];
        outp[i] = sums[i] / fmaxf(c, 1.0f);
    }
}

// ---------------------------------------------------------------------------
extern "C" void kernel_launch(void* const* d_in, const int* in_sizes, int n_in,
                              void* d_out, int out_size, void* d_ws, size_t ws_size,
                              hipStream_t stream) {
    const float* x     = (const float*)d_in[0];
    const int*   ei    = (const int*)  d_in[1];   // [2, E]
    const float* eattr = (const float*)d_in[2];   // [E, 3]
    const int*   batch = (const int*)  d_in[3];   // [N]
    const float* el1w  = (const float*)d_in[4];
    const float* el1b  = (const float*)d_in[5];
    const float* w1a   = (const float*)d_in[6];
    const float* b1a   = (const float*)d_in[7];
    const float* w1b   = (const float*)d_in[8];
    const float* b1b   = (const float*)d_in[9];
    const float* el2w  = (const float*)d_in[10];
    const float* el2b  = (const float*)d_in[11];
    const float* w2a   = (const float*)d_in[12];
    const float* b2a   = (const float*)d_in[13];
    const float* w2b   = (const float*)d_in[14];
    const float* b2b   = (const float*)d_in[15];

    const int* srcI = ei;
    const int* dstI = ei + N_EDGES_C;

    float* ws   = (float*)d_ws;
    float* agg  = ws;                                   // [N,128] (conv1 uses [N,64])
    float* h1   = agg  + (size_t)N_NODES_C * HID_C;     // [N,128]
    float* sums = h1   + (size_t)N_NODES_C * HID_C;     // [G,128]
    float* cnt  = sums + (size_t)N_GRAPHS_C * HID_C;    // [G]

    const int nodeBlocks = (N_NODES_C + 127) / 128;     // 391

    // ---- conv1 ----
    hipMemsetAsync(agg, 0, (size_t)N_NODES_C * IN_CH_C * sizeof(float), stream);
    {
        int total = N_EDGES_C * (IN_CH_C / 4);
        gine_edge_kernel<IN_CH_C><<<(total + 255) / 256, 256, 0, stream>>>(
            x, srcI, dstI, eattr, el1w, el1b, agg, N_EDGES_C);
    }
    gine_node_mlp_kernel<IN_CH_C><<<nodeBlocks, 256, 0, stream>>>(
        x, agg, w1a, b1a, w1b, b1b, h1, N_NODES_C);

    // ---- conv2 (node update runs in place over h1) ----
    hipMemsetAsync(agg, 0, (size_t)N_NODES_C * HID_C * sizeof(float), stream);
    {
        int total = N_EDGES_C * (HID_C / 4);
        gine_edge_kernel<HID_C><<<(total + 255) / 256, 256, 0, stream>>>(
            h1, srcI, dstI, eattr, el2w, el2b, agg, N_EDGES_C);
    }
    gine_node_mlp_kernel<HID_C><<<nodeBlocks, 256, 0, stream>>>(
        h1, agg, w2a, b2a, w2b, b2b, h1, N_NODES_C);

    // ---- global mean pool ----
    hipMemsetAsync(sums, 0, (size_t)(N_GRAPHS_C * HID_C + N_GRAPHS_C) * sizeof(float), stream);
    {
        int total = N_NODES_C * (HID_C / 4);
        pool_sum_kernel<<<(total + 255) / 256, 256, 0, stream>>>(
            h1, batch, sums, cnt, N_NODES_C);
    }
    pool_div_kernel<<<(N_GRAPHS_C * HID_C + 255) / 256, 256, 0, stream>>>(
        sums, cnt, (float*)d_out);
}